// SpatialTransformer_22565758173896
// MI455X (gfx1250) — compile-verified
//
#include <hip/hip_runtime.h>

// Batched 3D dense-warp spatial transformer (trilinear, edge-clamped), MI455X.
//
// vol: [2,128,192,192,1] f32, trf: [2,128,192,192,3] f32, out: [2,128,192,192,1] f32
//
// Memory-bound gather kernel (AI ~= 0.03 flop/byte): no matmul structure, so no
// WMMA path applies. Optimization targets the MI455X memory system instead:
//  - 4 voxels/thread -> trf read is 3x global_load_b128 (48B aligned run),
//    out write is 1x global_store_b128.
//  - trf/out marked non-temporal (TH=NT) so the 113MB trf stream does not
//    evict the 38MB vol working set from the 192MB L2; vol gathers (default RT)
//    then hit L2 after one cold pass.
//  - global_prefetch_b8 warms L2 ~36KB ahead of the trf stream.

typedef float v4f __attribute__((ext_vector_type(4)));

#define BD 2
#define DD 128
#define HH 192
#define WW 192
#define S0 (HH * WW)        // z stride = 36864
#define S1 (WW)             // y stride = 192
#define VOLN (DD * HH * WW) // per-batch voxels = 4718592

__global__ __launch_bounds__(192) void warp3d_trilinear_kernel(
    const float* __restrict__ vol,
    const float* __restrict__ trf,
    float* __restrict__ out)
{
    // block: (48, 4, 1) -> threadIdx.x covers x in steps of 4, threadIdx.y covers 4 rows
    // grid : (H/4, D, B)
    const int x4 = threadIdx.x << 2;                     // 0..188, step 4
    const int y  = (blockIdx.x << 2) + threadIdx.y;      // 0..191
    const int z  = blockIdx.y;                           // 0..127
    const int b  = blockIdx.z;                           // 0..1

    const int voxBase = ((b * DD + z) * HH + y) * WW + x4;

    // ---- streaming trf read: 12 floats = 3 aligned float4, non-temporal ----
    const float* t = trf + (size_t)voxBase * 3;
    // warm L2 ahead of the stream (speculative global_prefetch_b8; OOB is dropped)
    __builtin_prefetch(t + 9216, 0, 1);
    const v4f t0 = __builtin_nontemporal_load((const v4f*)(t + 0));
    const v4f t1 = __builtin_nontemporal_load((const v4f*)(t + 4));
    const v4f t2 = __builtin_nontemporal_load((const v4f*)(t + 8));

    // per-voxel displacement components: trf[...,0]->z, [...,1]->y, [...,2]->x
    const float dz[4] = { t0.x, t0.w, t1.z, t2.y };
    const float dy[4] = { t0.y, t1.x, t1.w, t2.z };
    const float dx[4] = { t0.z, t1.y, t2.x, t2.w };

    const float* __restrict__ vb = vol + (size_t)b * VOLN;

    float res[4];
#pragma unroll
    for (int k = 0; k < 4; ++k) {
        // absolute sample coordinate, edge-clamped (voxelmorph fill_value=None)
        const float cz = fminf(fmaxf((float)z        + dz[k], 0.0f), (float)(DD - 1));
        const float cy = fminf(fmaxf((float)y        + dy[k], 0.0f), (float)(HH - 1));
        const float cx = fminf(fmaxf((float)(x4 + k) + dx[k], 0.0f), (float)(WW - 1));

        const int z0 = (int)floorf(cz);
        const int y0 = (int)floorf(cy);
        const int x0 = (int)floorf(cx);
        const int z1 = min(z0 + 1, DD - 1);
        const int y1 = min(y0 + 1, HH - 1);
        const int x1 = min(x0 + 1, WW - 1);

        // low-corner weights d1 = loc1 - cloc; high-corner weights d0 = 1 - d1
        const float lz = (float)z1 - cz, hz = 1.0f - lz;
        const float ly = (float)y1 - cy, hy = 1.0f - ly;
        const float lx = (float)x1 - cx, hx = 1.0f - lx;

        // 8-corner gather; displacements are ~N(0,1) so corners are L0/L2-local
        const float* p00 = vb + z0 * S0 + y0 * S1;
        const float* p01 = vb + z0 * S0 + y1 * S1;
        const float* p10 = vb + z1 * S0 + y0 * S1;
        const float* p11 = vb + z1 * S0 + y1 * S1;
        const float v000 = p00[x0], v001 = p00[x1];
        const float v010 = p01[x0], v011 = p01[x1];
        const float v100 = p10[x0], v101 = p10[x1];
        const float v110 = p11[x0], v111 = p11[x1];

        // factored trilinear lerp: 7 fma + 7 mul
        const float c00 = fmaf(v000, lx, v001 * hx);
        const float c01 = fmaf(v010, lx, v011 * hx);
        const float c10 = fmaf(v100, lx, v101 * hx);
        const float c11 = fmaf(v110, lx, v111 * hx);
        const float c0  = fmaf(c00, ly, c01 * hy);
        const float c1  = fmaf(c10, ly, c11 * hy);
        res[k] = fmaf(c0, lz, c1 * hz);
    }

    // ---- streaming out write: one aligned non-temporal b128 store ----
    v4f o;
    o.x = res[0]; o.y = res[1]; o.z = res[2]; o.w = res[3];
    __builtin_nontemporal_store(o, (v4f*)(out + voxBase));
}

extern "C" void kernel_launch(void* const* d_in, const int* in_sizes, int n_in,
                              void* d_out, int out_size, void* d_ws, size_t ws_size,
                              hipStream_t stream) {
    const float* vol = (const float*)d_in[0];
    const float* trf = (const float*)d_in[1];
    float* out = (float*)d_out;

    dim3 block(48, 4, 1);          // 192 threads = 6 wave32
    dim3 grid(HH / 4, DD, BD);     // (48, 128, 2) blocks
    warp3d_trilinear_kernel<<<grid, block, 0, stream>>>(vol, trf, out);
}